// MedianFilter1D_30812095382121
// MI455X (gfx1250) — compile-verified
//
#include <hip/hip_runtime.h>

#define L_LEN 8192
#define TILE  1024
#define TILES_PER_ROW (L_LEN / TILE)   // 8 (power of two -> no integer division)
#define HALO  4                        // (k-1)/2 with k=9

// 128-bit payload type for the async global->LDS DMA builtin
// (clang's param type: int __attribute__((vector_size(16))) in AS1 / AS3)
typedef int v4i_ __attribute__((vector_size(4 * sizeof(int))));
typedef v4i_ __attribute__((address_space(1))) as1_v4i;
typedef v4i_ __attribute__((address_space(3))) as3_v4i;

// Paeth's optimal median-of-9 network: 19 compare-exchanges = 38 min/max VALU ops.
// Exact order statistic (k odd), branch-free, maps to v_min_num_f32/v_max_num_f32.
__device__ __forceinline__ float median9(float p0, float p1, float p2,
                                         float p3, float p4, float p5,
                                         float p6, float p7, float p8) {
#define S2(a, b) { float _t = fminf(a, b); b = fmaxf(a, b); a = _t; }
    S2(p1, p2) S2(p4, p5) S2(p7, p8)
    S2(p0, p1) S2(p3, p4) S2(p6, p7)
    S2(p1, p2) S2(p4, p5) S2(p7, p8)
    S2(p0, p3) S2(p5, p8) S2(p4, p7)
    S2(p3, p6) S2(p1, p4) S2(p2, p5)
    S2(p4, p7) S2(p4, p2) S2(p6, p4)
    S2(p4, p2)
#undef S2
    return p4;
}

__global__ __launch_bounds__(256) void median_filter1d_kernel(
    const float* __restrict__ x, float* __restrict__ y) {
    __shared__ __align__(16) float smem[TILE + 2 * HALO];

    const int tid        = threadIdx.x;
    const int tile       = (int)(blockIdx.x & (TILES_PER_ROW - 1));
    const int row        = (int)(blockIdx.x >> 3);  // / TILES_PER_ROW
    const size_t rowBase = (size_t)row * (size_t)L_LEN;
    const int tile_start = tile * TILE;
    const float* rowPtr  = x + rowBase;

    // ---- Bulk tile: 256 lanes x 16B = 4096B, async DMA into LDS ----
    {
        float* gsrc = const_cast<float*>(rowPtr + tile_start + 4 * tid); // 16B aligned
        float* ldst = &smem[HALO + 4 * tid];                             // 16B aligned
#if defined(__gfx1250__) && __has_builtin(__builtin_amdgcn_global_load_async_to_lds_b128)
        __builtin_amdgcn_global_load_async_to_lds_b128(
            (as1_v4i*)gsrc,
            (as3_v4i*)ldst,
            /*offset=*/0, /*cpol=*/0);
#else
        *reinterpret_cast<float4*>(ldst) = *reinterpret_cast<const float4*>(gsrc);
#endif
    }

    // ---- Halo: 8 clamped scalar loads implement replicate padding ----
    if (tid < 2 * HALO) {
        int g, s;
        if (tid < HALO) { g = tile_start - HALO + tid;        s = tid; }
        else            { g = tile_start + TILE + tid - HALO; s = TILE + HALO + tid - HALO; }
        g = g < 0 ? 0 : (g > (L_LEN - 1) ? (L_LEN - 1) : g);
        smem[s] = rowPtr[g];
    }

#if defined(__gfx1250__) && __has_builtin(__builtin_amdgcn_global_load_async_to_lds_b128)
#  if __has_builtin(__builtin_amdgcn_s_wait_asynccnt)
    __builtin_amdgcn_s_wait_asynccnt(0);
#  else
    asm volatile("s_wait_asynccnt 0" ::: "memory");
#  endif
#endif
    __syncthreads();

    // ---- Compute: 12 LDS floats (3x ds_load_b128) -> 4 medians ----
    const float4 a = *reinterpret_cast<const float4*>(&smem[4 * tid]);
    const float4 b = *reinterpret_cast<const float4*>(&smem[4 * tid + 4]);
    const float4 c = *reinterpret_cast<const float4*>(&smem[4 * tid + 8]);
    const float w0 = a.x, w1 = a.y, w2 = a.z, w3 = a.w;
    const float w4 = b.x, w5 = b.y, w6 = b.z, w7 = b.w;
    const float w8 = c.x, w9 = c.y, w10 = c.z, w11 = c.w;

    float4 o;
    o.x = median9(w0, w1, w2, w3, w4, w5, w6, w7, w8);
    o.y = median9(w1, w2, w3, w4, w5, w6, w7, w8, w9);
    o.z = median9(w2, w3, w4, w5, w6, w7, w8, w9, w10);
    o.w = median9(w3, w4, w5, w6, w7, w8, w9, w10, w11);

    *reinterpret_cast<float4*>(y + rowBase + tile_start + 4 * tid) = o;
}

extern "C" void kernel_launch(void* const* d_in, const int* in_sizes, int n_in,
                              void* d_out, int out_size, void* d_ws, size_t ws_size,
                              hipStream_t stream) {
    (void)n_in; (void)out_size; (void)d_ws; (void)ws_size;
    const float* x = (const float*)d_in[0];
    float* y       = (float*)d_out;
    // kernel_size (d_in[1]) is statically 9 per the reference.
    const int total = in_sizes[0];                     // 32*64*8192
    const int rows  = total / L_LEN;                   // 2048
    dim3 grid((unsigned)(rows * TILES_PER_ROW));       // 16384 blocks
    median_filter1d_kernel<<<grid, 256, 0, stream>>>(x, y);
}